// GNNRankModel_33122787786914
// MI455X (gfx1250) — compile-verified
//
#include <hip/hip_runtime.h>
#include <hip/hip_bf16.h>

typedef __attribute__((ext_vector_type(2))) float v2f;
typedef __attribute__((ext_vector_type(8))) float v8f;

#define D 128

// ---------------------------------------------------------------------------
// Degree accumulation: deg_out[src[e]] += 1 ; deg_in[dst[e]] += 1
// ---------------------------------------------------------------------------
__global__ __launch_bounds__(256) void gcn_degree(const int* __restrict__ es,
                                                  const int* __restrict__ ed,
                                                  float* __restrict__ dout,
                                                  float* __restrict__ din,
                                                  int E) {
  int i = blockIdx.x * blockDim.x + threadIdx.x;
  if (i < E) {
    atomicAdd(&dout[es[i]], 1.0f);
    atomicAdd(&din[ed[i]], 1.0f);
  }
}

// deg -> rsqrt(max(deg,1)) in place, over 2N contiguous floats
__global__ __launch_bounds__(256) void gcn_rsqrt(float* __restrict__ deg, int n) {
  int i = blockIdx.x * blockDim.x + threadIdx.x;
  if (i < n) {
    float d = deg[i];
    d = d < 1.0f ? 1.0f : d;
    deg[i] = rsqrtf(d);
  }
}

// out[n][:] = in[n][:] * r[n]   (float4 per thread; n4 = N*32 float4s)
__global__ __launch_bounds__(256) void gcn_scale(const float* __restrict__ in,
                                                 const float* __restrict__ r,
                                                 float* __restrict__ out,
                                                 int n4) {
  int i = blockIdx.x * blockDim.x + threadIdx.x;
  if (i >= n4) return;
  int row = i >> 5;  // 32 float4 per 128-wide row
  float rs = r[row];
  float4 v = ((const float4*)in)[i];
  v.x *= rs; v.y *= rs; v.z *= rs; v.w *= rs;
  ((float4*)out)[i] = v;
}

// ---------------------------------------------------------------------------
// SpMM scatter: accum[dst[e]][:] += feat[src[e]][:]
// One wave32 per edge, 4 contiguous floats per lane (coalesced 512B/row).
// ---------------------------------------------------------------------------
__global__ __launch_bounds__(256) void gcn_scatter(const float* __restrict__ feat,
                                                   const int* __restrict__ es,
                                                   const int* __restrict__ ed,
                                                   float* __restrict__ accum,
                                                   int E) {
  long long gid = (long long)blockIdx.x * blockDim.x + threadIdx.x;
  int e = (int)(gid >> 5);
  if (e >= E) return;
  int c = ((int)gid & 31) * 4;
  int s = es[e];
  int d = ed[e];
  const float4 v = *(const float4*)(feat + (size_t)s * D + c);
  float* dp = accum + (size_t)d * D + c;
  atomicAdd(dp + 0, v.x);
  atomicAdd(dp + 1, v.y);
  atomicAdd(dp + 2, v.z);
  atomicAdd(dp + 3, v.w);
}

// ---------------------------------------------------------------------------
// Fused GEMM: Out = relu( (Min * rowscale[row]) @ W + bias ), all fp32 via
// V_WMMA_F32_16X16X4_F32. One wave owns a 16-row strip x 128 cols
// (8 accumulators of v8f). W (64KB) staged in LDS in a (k-pair, col)
// interleaved layout so each B fragment is a single ds_load_b64:
//   lds[(k>>1)*256 + col*2 + (k&1)] = W[k][col]
// ---------------------------------------------------------------------------
__global__ __launch_bounds__(256) void gcn_gemm_wmma(
    const float* __restrict__ Min,       // [N, 128]
    const float* __restrict__ rowscale,  // [N]  (r_in, fused dst-normalization)
    const float* __restrict__ W,         // [128, 128] row-major (k, col)
    const float* __restrict__ bias,      // [128]
    float* __restrict__ Out,             // [N, 128]
    int nrows) {
  __shared__ float ldsW[D * D];  // 64 KB

  const int t = threadIdx.x;
  // Stage W -> LDS (coalesced float4 global loads, scalar ds stores)
  for (int i = 0; i < 16; ++i) {
    int e4 = t + i * 256;          // 4096 float4 total
    int k = e4 >> 5;               // 32 float4 per k-row
    int c4 = (e4 & 31) * 4;
    const float4 w4 = ((const float4*)W)[e4];
    int base = (k >> 1) * 256 + (k & 1);
    ldsW[base + (c4 + 0) * 2] = w4.x;
    ldsW[base + (c4 + 1) * 2] = w4.y;
    ldsW[base + (c4 + 2) * 2] = w4.z;
    ldsW[base + (c4 + 3) * 2] = w4.w;
  }
  __syncthreads();

  const int wave = t >> 5;
  const int lane = t & 31;
  const int tile = blockIdx.x * 8 + wave;  // 16-row output tile (wave-uniform)
  if (tile * 16 >= nrows) return;          // uniform branch: EXEC stays all-1s

  const int half = lane >> 4;   // 0: K=0,1 ; 1: K=2,3  (A and B fragments)
  const int l16  = lane & 15;
  const int row  = tile * 16 + l16;   // A-matrix M row for this lane
  const int koff = half * 2;
  const float rs = rowscale[row];
  const float* arow = Min + (size_t)row * D;
  __builtin_prefetch(arow, 0, 3);  // global_prefetch_b8

  v8f acc[8];
#pragma unroll
  for (int nt = 0; nt < 8; ++nt) acc[nt] = (v8f){0, 0, 0, 0, 0, 0, 0, 0};

#pragma unroll 4
  for (int k0 = 0; k0 < D; k0 += 4) {
    v2f a = *(const v2f*)(arow + k0 + koff);  // 8B aligned (k0+koff even)
    a.x *= rs;
    a.y *= rs;
    const int kb = (k0 + koff) >> 1;  // pair index into staged W
#pragma unroll
    for (int nt = 0; nt < 8; ++nt) {
      const int col = nt * 16 + l16;
      v2f b = *(const v2f*)&ldsW[kb * 256 + col * 2];  // ds_load_b64
      acc[nt] = __builtin_amdgcn_wmma_f32_16x16x4_f32(
          false, a, false, b, (short)0, acc[nt], false, false);
    }
  }

  // Epilogue: acc[nt][r] -> row tile*16 + r + half*8, col nt*16 + l16
  const int orow = tile * 16 + half * 8;
#pragma unroll
  for (int nt = 0; nt < 8; ++nt) {
    const int col = nt * 16 + l16;
    const float bv = bias[col];
#pragma unroll
    for (int r = 0; r < 8; ++r) {
      float v = acc[nt][r] + bv;
      Out[(size_t)(orow + r) * D + col] = v > 0.0f ? v : 0.0f;
    }
  }
}

// ---------------------------------------------------------------------------
// Edge scoring: out[e] = dot(h[ss[e]], h[sd[e]]) over 128 dims.
// Wave32 per edge, float4 per lane, shfl_xor reduction.
// ---------------------------------------------------------------------------
__global__ __launch_bounds__(256) void gcn_score(const float* __restrict__ h,
                                                 const int* __restrict__ ss,
                                                 const int* __restrict__ sd,
                                                 float* __restrict__ out,
                                                 int ES) {
  long long gid = (long long)blockIdx.x * blockDim.x + threadIdx.x;
  int e = (int)(gid >> 5);
  if (e >= ES) return;
  int lane = (int)gid & 31;
  int c = lane * 4;
  const float4 a = *(const float4*)(h + (size_t)ss[e] * D + c);
  const float4 b = *(const float4*)(h + (size_t)sd[e] * D + c);
  float s = a.x * b.x + a.y * b.y + a.z * b.z + a.w * b.w;
#pragma unroll
  for (int off = 16; off > 0; off >>= 1) s += __shfl_xor(s, off, 32);
  if (lane == 0) out[e] = s;
}

// ---------------------------------------------------------------------------
extern "C" void kernel_launch(void* const* d_in, const int* in_sizes, int n_in,
                              void* d_out, int out_size, void* d_ws, size_t ws_size,
                              hipStream_t stream) {
  const float* x  = (const float*)d_in[0];
  const int* es   = (const int*)d_in[1];
  const int* ed   = (const int*)d_in[2];
  const int* ss   = (const int*)d_in[3];
  const int* sd   = (const int*)d_in[4];
  const float* W1 = (const float*)d_in[5];
  const float* b1 = (const float*)d_in[6];
  const float* W2 = (const float*)d_in[7];
  const float* b2 = (const float*)d_in[8];

  const int N  = in_sizes[0] / D;  // 50000
  const int E  = in_sizes[1];      // 800000
  const int ES = in_sizes[3];      // 200000
  float* out = (float*)d_out;

  // Workspace: [deg_out|deg_in (2N)] [bufA N*D] [bufB N*D] [bufC N*D]
  float* deg   = (float*)d_ws;
  float* r_out = deg;
  float* r_in  = deg + N;
  float* bufA  = deg + 2 * (size_t)N;
  float* bufB  = bufA + (size_t)N * D;
  float* bufC  = bufB + (size_t)N * D;

  const int n4 = N * (D / 4);                              // float4s per feature buf
  const long long nscat = (long long)E * 32;               // scatter threads
  const int scatBlocks  = (int)((nscat + 255) / 256);
  const int gemmBlocks  = ((N + 15) / 16 + 7) / 8;         // 16-row tiles, 8 waves/blk
  const long long nscore = (long long)ES * 32;
  const int scoreBlocks  = (int)((nscore + 255) / 256);

  // Degrees -> rsqrt(clip(deg,1))
  hipMemsetAsync(deg, 0, sizeof(float) * 2 * (size_t)N, stream);
  gcn_degree<<<(E + 255) / 256, 256, 0, stream>>>(es, ed, r_out, r_in, E);
  gcn_rsqrt<<<(2 * N + 255) / 256, 256, 0, stream>>>(deg, 2 * N);

  // Layer 1
  gcn_scale<<<(n4 + 255) / 256, 256, 0, stream>>>(x, r_out, bufA, n4);
  hipMemsetAsync(bufB, 0, sizeof(float) * (size_t)N * D, stream);
  gcn_scatter<<<scatBlocks, 256, 0, stream>>>(bufA, es, ed, bufB, E);
  gcn_gemm_wmma<<<gemmBlocks, 256, 0, stream>>>(bufB, r_in, W1, b1, bufC, N);

  // Layer 2
  gcn_scale<<<(n4 + 255) / 256, 256, 0, stream>>>(bufC, r_out, bufA, n4);
  hipMemsetAsync(bufB, 0, sizeof(float) * (size_t)N * D, stream);
  gcn_scatter<<<scatBlocks, 256, 0, stream>>>(bufA, es, ed, bufB, E);
  gcn_gemm_wmma<<<gemmBlocks, 256, 0, stream>>>(bufB, r_in, W2, b2, bufC, N);

  // Edge scores
  gcn_score<<<scoreBlocks, 256, 0, stream>>>(bufC, ss, sd, out, ES);
}